// Attention_67869073211704
// MI455X (gfx1250) — compile-verified
//
#include <hip/hip_runtime.h>
#include <cmath>

// Problem dims (fixed by the reference harness)
constexpr int Bn = 128;   // batch
constexpr int Ln = 2048;  // sequence length
constexpr int Hn = 512;   // hidden

// k2 tiling: full-K A panel in LDS (CDNA5: 320KB LDS per WGP)
constexpr int ROWS = 64;  // L rows per block (4 M-tiles of 16)
constexpr int AS   = 516; // padded LDS row stride: bank = (4m + k) % 64 -> conflict-free

typedef float v2f __attribute__((ext_vector_type(2)));
typedef float v8f __attribute__((ext_vector_type(8)));

// ---------------------------------------------------------------------------
// k1: qb[b,h] = query[b,:]·Wq[h,:] + bq[h] + br[h]   (both biases folded pre-tanh)
// ---------------------------------------------------------------------------
__global__ void k1_qproj(const float* __restrict__ query, const float* __restrict__ Wq,
                         const float* __restrict__ bq, const float* __restrict__ br,
                         float* __restrict__ qb) {
    int gid = blockIdx.x * blockDim.x + threadIdx.x;   // B*H threads
    int b = gid >> 9;          // /512
    int h = gid & (Hn - 1);
    const float4* q4 = (const float4*)(query + (size_t)b * Hn);
    const float4* w4 = (const float4*)(Wq + (size_t)h * Hn);
    float s = 0.f;
#pragma unroll 4
    for (int k = 0; k < Hn / 4; ++k) {
        float4 a = q4[k], w = w4[k];
        s = fmaf(a.x, w.x, s); s = fmaf(a.y, w.y, s);
        s = fmaf(a.z, w.z, s); s = fmaf(a.w, w.w, s);
    }
    qb[gid] = s + bq[h] + br[h];
}

// ---------------------------------------------------------------------------
// k2: fused  scores[b,l] = vw · tanh(qb[b,:] + ref[b,l,:]·Wr^T) + vb  (+ mask)
// Block = (b, 64 L-rows), 8 waves. Full 64x512 A panel staged once in LDS
// (132 KB -- CDNA5 large LDS). Wave loop: t = 4 N-tiles (outer), holding only
// acc[4 M-tiles] (32 VGPRs) live -> no spills. Per k-step:
//   1 global_load_b64 (Wr frag, reused by 4 WMMAs) + 4 ds_load_b64 + 4 WMMAs.
// r_proj is never written to memory.
// ---------------------------------------------------------------------------
__global__ void k2_scores(const float* __restrict__ ref, const float* __restrict__ Wr,
                          const float* __restrict__ qb, const float* __restrict__ vw,
                          const float* __restrict__ vb, const int* __restrict__ mask,
                          float* __restrict__ scores) {
    __shared__ float Atile[ROWS * AS];    // 64 x 512 (padded) = 132 KB
    __shared__ float qb_s[Hn];
    __shared__ float vw_s[Hn];
    __shared__ float score_s[ROWS];

    const int b   = blockIdx.y;
    const int l0  = blockIdx.x * ROWS;
    const int tid = threadIdx.x;

    const int wave = tid >> 5;
    const int lane = tid & 31;
    const int half = lane >> 4;   // 0: K pair {0,1}, 1: K pair {2,3}
    const int nl   = lane & 15;

    for (int i = tid; i < Hn; i += 256) { qb_s[i] = qb[(size_t)b * Hn + i]; vw_s[i] = vw[i]; }
    if (tid < ROWS) score_s[tid] = 0.f;

    // cooperative, coalesced float4 load of the full 64 x 512 A panel (once)
    const float* refb = ref + ((size_t)b * Ln + l0) * Hn;
    for (int idx = tid; idx < ROWS * (Hn / 4); idx += 256) {
        int row = idx >> 7;            // /(Hn/4 = 128)
        int c4  = (idx & 127) << 2;
        float4 v = *(const float4*)(refb + (size_t)row * Hn + c4);
        float* dst = &Atile[row * AS + c4];
        dst[0] = v.x; dst[1] = v.y; dst[2] = v.z; dst[3] = v.w;
    }
    __syncthreads();

    for (int t = 0; t < 4; ++t) {
        const int n = (wave + t * 8) * 16 + nl;      // this lane's output column
        const float* wrow = Wr + (size_t)n * Hn;     // B frag: B[k][n] = Wr[n][k]

        v8f acc[4] = {};   // 4 M-tiles, 32 VGPRs live

#pragma unroll 2
        for (int k0 = 0; k0 < Hn; k0 += 4) {
            const int k = k0 + 2 * half;
            v2f bb;                                   // 1 global_load_b64 (L2-resident Wr)
            bb.x = wrow[k]; bb.y = wrow[k + 1];
#pragma unroll
            for (int mt = 0; mt < 4; ++mt) {          // 4 ds_load_b64 + 4 independent WMMAs
                const float* ar = &Atile[(mt * 16 + nl) * AS + k];
                v2f a; a.x = ar[0]; a.y = ar[1];
                acc[mt] = __builtin_amdgcn_wmma_f32_16x16x4_f32(
                    false, a, false, bb, (short)0, acc[mt], false, false);
            }
        }

        // epilogue: tv = vw[n]*tanh(d + qb[n]); reduce over the 16 N-lanes per row
        const float qv = qb_s[n];
        const float wv = vw_s[n];
#pragma unroll
        for (int mt = 0; mt < 4; ++mt) {
#pragma unroll
            for (int v = 0; v < 8; ++v) {
                // D layout: vgpr v, lanes 0-15 -> M=v, lanes 16-31 -> M=v+8 (N = lane%16)
                float tv = wv * tanhf(acc[mt][v] + qv);
                tv += __shfl_xor(tv, 1, 32);
                tv += __shfl_xor(tv, 2, 32);
                tv += __shfl_xor(tv, 4, 32);
                tv += __shfl_xor(tv, 8, 32);
                if (nl == 0) atomicAdd(&score_s[mt * 16 + v + 8 * half], tv); // ds_add_f32
            }
        }
    }
    __syncthreads();

    if (tid < ROWS) {
        const int l = l0 + tid;
        float s = score_s[tid] + vb[0];
        if (mask[(size_t)b * Ln + l] == 0) s = -1e9f;
        scores[(size_t)b * Ln + l] = s;
    }
}

// ---------------------------------------------------------------------------
// k3: softmax over L per batch row (block per batch)
// ---------------------------------------------------------------------------
__global__ void k3_softmax(const float* __restrict__ scores, float* __restrict__ attn) {
    __shared__ float red[256];
    const int b = blockIdx.x, tid = threadIdx.x;
    const float* srow = scores + (size_t)b * Ln;
    float* arow = attn + (size_t)b * Ln;

    float m = -3.402823466e38f;
    for (int l = tid; l < Ln; l += 256) m = fmaxf(m, srow[l]);
    red[tid] = m; __syncthreads();
    for (int s = 128; s > 0; s >>= 1) { if (tid < s) red[tid] = fmaxf(red[tid], red[tid + s]); __syncthreads(); }
    m = red[0]; __syncthreads();

    float sum = 0.f;
    for (int l = tid; l < Ln; l += 256) { float e = __expf(srow[l] - m); arow[l] = e; sum += e; }
    red[tid] = sum; __syncthreads();
    for (int s = 128; s > 0; s >>= 1) { if (tid < s) red[tid] += red[tid + s]; __syncthreads(); }
    const float inv = 1.f / red[0];
    for (int l = tid; l < Ln; l += 256) arow[l] *= inv;
}

// ---------------------------------------------------------------------------
// k4: context[b,h] = sum_l attn[b,l] * ref[b,l,h]  — the bandwidth-dominant pass.
// Block = (b, 256 consecutive h). attn chunk staged in LDS, ref reads coalesced.
// ---------------------------------------------------------------------------
__global__ void k4_context(const float* __restrict__ attn, const float* __restrict__ ref,
                           float* __restrict__ ctx) {
    __shared__ float w_s[256];
    const int b = blockIdx.y;
    const int h = blockIdx.x * 256 + threadIdx.x;
    const float* refb = ref + (size_t)b * Ln * Hn;
    const float* arow = attn + (size_t)b * Ln;
    float acc = 0.f;
    for (int l0 = 0; l0 < Ln; l0 += 256) {
        __syncthreads();
        w_s[threadIdx.x] = arow[l0 + threadIdx.x];
        __syncthreads();
        if (l0 + 256 < Ln) __builtin_prefetch(&refb[(size_t)(l0 + 256) * Hn + h], 0, 0);
#pragma unroll 8
        for (int j = 0; j < 256; ++j)
            acc = fmaf(w_s[j], refb[(size_t)(l0 + j) * Hn + h], acc);
    }
    ctx[(size_t)b * Hn + h] = acc;
}

// ---------------------------------------------------------------------------
extern "C" void kernel_launch(void* const* d_in, const int* in_sizes, int n_in,
                              void* d_out, int out_size, void* d_ws, size_t ws_size,
                              hipStream_t stream) {
    const float* query = (const float*)d_in[0];
    const float* ref   = (const float*)d_in[1];
    const int*   mask  = (const int*)  d_in[2];
    const float* Wq    = (const float*)d_in[3];
    const float* bq    = (const float*)d_in[4];
    const float* Wr    = (const float*)d_in[5];
    const float* br    = (const float*)d_in[6];
    const float* vw    = (const float*)d_in[7];
    const float* vb    = (const float*)d_in[8];

    float* ctx  = (float*)d_out;                   // [B,H]
    float* attn = (float*)d_out + (size_t)Bn * Hn; // [B,L]

    float* qb     = (float*)d_ws;                  // [B,H]
    float* scores = qb + (size_t)Bn * Hn;          // [B,L]

    k1_qproj  <<<(Bn * Hn) / 256, 256, 0, stream>>>(query, Wq, bq, br, qb);
    k2_scores <<<dim3(Ln / ROWS, Bn), 256, 0, stream>>>(ref, Wr, qb, vw, vb, mask, scores);
    k3_softmax<<<Bn, 256, 0, stream>>>(scores, attn);
    k4_context<<<dim3(Hn / 256, Bn), 256, 0, stream>>>(attn, ref, ctx);
}